// OpticalNetworkEncoder_8203387535414
// MI455X (gfx1250) — compile-verified
//
#include <hip/hip_runtime.h>
#include <math.h>
#include <stdint.h>

// Problem dims (match reference)
constexpr int Bc = 8, Nc = 2048, Ec = 8192, Sc = 80, LPc = 256;
constexpr int Dc = 128, Hc = 4, SPDc = 64, LPDc = 32, GDc = 8, LATc = 128;
constexpr int NLc = 3;
constexpr int BEc = Bc * Ec;   // 65536 edges total
constexpr int BNc = Bc * Nc;   // 16384 nodes total

typedef __attribute__((ext_vector_type(2))) float v2f;
typedef __attribute__((ext_vector_type(8))) float v8f;

// ---------------------------------------------------------------------------
// CDNA5 async global->LDS copies (ASYNCcnt-tracked). The low 32 bits of a flat
// pointer into __shared__ are the LDS byte offset (ISA 10.2 aperture mapping).
// ---------------------------------------------------------------------------
__device__ __forceinline__ void async_g2l_b128(unsigned lds_byte, const float* gaddr) {
  asm volatile("global_load_async_to_lds_b128 %0, %1, off"
               :: "v"(lds_byte), "v"(gaddr) : "memory");
}
__device__ __forceinline__ void async_g2l_b32(unsigned lds_byte, const float* gaddr) {
  asm volatile("global_load_async_to_lds_b32 %0, %1, off"
               :: "v"(lds_byte), "v"(gaddr) : "memory");
}
__device__ __forceinline__ void wait_asynccnt0() {
  asm volatile("s_wait_asynccnt 0x0" ::: "memory");
}
__device__ __forceinline__ void wait_asynccnt3() {
  asm volatile("s_wait_asynccnt 0x3" ::: "memory");
}

// ---------------------------------------------------------------------------
// Generic fill
// ---------------------------------------------------------------------------
__global__ void k_fill(float* __restrict__ p, float v, size_t n) {
  size_t i = (size_t)blockIdx.x * blockDim.x + threadIdx.x;
  if (i < n) p[i] = v;
}

// Bt[k][n] = W[n][k] for k < K0 else 0 ; Bt is [Kp x Nrows]
__global__ void k_transpose_pad(const float* __restrict__ W, float* __restrict__ Bt,
                                int Nrows, int K0, int Kp) {
  int idx = blockIdx.x * blockDim.x + threadIdx.x;
  int total = Kp * Nrows;
  if (idx >= total) return;
  int k = idx / Nrows, n = idx - k * Nrows;
  Bt[idx] = (k < K0) ? W[(size_t)n * K0 + k] : 0.0f;
}

// ---------------------------------------------------------------------------
// WMMA fp32 GEMM with double-buffered async-LDS staging.
// One block (8 waves / 256 threads) computes a 16-row x 128-col output strip.
// K chunked by 16. Per chunk each thread issues EXACTLY 3 async copies
// (1x b32 for A, 2x b128 for B) so ASYNCcnt bookkeeping is exact: after
// issuing chunk c+1, s_wait_asynccnt 3 guarantees chunk c landed (async loads
// complete in order, ISA 08 sect.4.1). Each wave then runs 4x
// V_WMMA_F32_16X16X4_F32 on its 16x16 tile, operands read from LDS.
// A 16x4 lane layout (ISA 7.12.2): lanes 0-15 hold K={k0,k0+1}, 16-31 K={k0+2,k0+3}.
// ---------------------------------------------------------------------------
#define KC 16
__device__ __forceinline__ void stage_chunk(
    float (*As)[20], float (*Bs)[128],
    const float* A, int lda, const float* Bt, int tm, int k0, int tid)
{
  // A tile 16x16: one b32 per thread
  int row = tid >> 4, seg = tid & 15;
  async_g2l_b32((unsigned)(size_t)&As[row][seg],
                A + (size_t)(tm * 16 + row) * lda + k0 + seg);
  // B chunk 16x128: two b128 per thread (512 segments total)
  int j0 = tid * 2;
  int k1 = j0 >> 5, s1 = j0 & 31;
  async_g2l_b128((unsigned)(size_t)&Bs[k1][s1 * 4],
                 Bt + (size_t)(k0 + k1) * 128 + s1 * 4);
  int j1 = j0 + 1;
  int k2 = j1 >> 5, s2 = j1 & 31;
  async_g2l_b128((unsigned)(size_t)&Bs[k2][s2 * 4],
                 Bt + (size_t)(k0 + k2) * 128 + s2 * 4);
}

__global__ void __launch_bounds__(256) k_wmma_gemm(
    const float* __restrict__ A, int lda,
    const float* __restrict__ Bt,            // [Kdim x 128]
    const float* __restrict__ bias,
    const float* __restrict__ addsrc,
    float* __restrict__ C,                   // [M x 128]
    int Kdim, int relu)
{
  __shared__ float As[2][16][20];            // padded stride 20 floats (bank spread)
  __shared__ float Bs[2][KC][128];
  int tm   = blockIdx.x;                     // 16-row strip index
  int tid  = threadIdx.x;
  int wave = tid >> 5;
  int lane = tid & 31;
  int ln   = lane & 15;
  int sel  = lane >> 4;
  int cw   = wave * 16 + ln;                 // this wave's output column for ln

  int nch = Kdim / KC;
  stage_chunk(As[0], Bs[0], A, lda, Bt, tm, 0, tid);   // prologue: chunk 0 in flight

  v8f acc = {};
  for (int c = 0; c < nch; ++c) {
    if (c + 1 < nch) {
      // overlap: issue chunk c+1 into the other buffer, then wait only for
      // chunk c (the 3 copies issued one iteration ago)
      stage_chunk(As[(c + 1) & 1], Bs[(c + 1) & 1], A, lda, Bt, tm, (c + 1) * KC, tid);
      wait_asynccnt3();
    } else {
      wait_asynccnt0();
    }
    __syncthreads();                         // everyone's chunk-c copies landed
    const float (*Asb)[20]  = As[c & 1];
    const float (*Bsb)[128] = Bs[c & 1];
#pragma unroll
    for (int kk = 0; kk < KC; kk += 4) {
      int ka = kk + 2 * sel;
      v2f a = *(const v2f*)&Asb[ln][ka];
      v2f b;
      b.x = Bsb[ka][cw];
      b.y = Bsb[ka + 1][cw];
      acc = __builtin_amdgcn_wmma_f32_16x16x4_f32(
          false, a, false, b, (short)0, acc, false, false);
    }
    __syncthreads();                         // chunk-c buffer free for re-stage
  }

  float bv = bias ? bias[cw] : 0.0f;
#pragma unroll
  for (int r = 0; r < 8; ++r) {
    int orow = tm * 16 + r + 8 * sel;        // C layout: VGPR r -> M=r (lanes 0-15), M=r+8 (16-31)
    float v = acc[r] + bv;
    if (addsrc) v += addsrc[(size_t)orow * 128 + cw];
    if (relu) v = fmaxf(v, 0.0f);
    C[(size_t)orow * 128 + cw] = v;
  }
}

// ---------------------------------------------------------------------------
// Spectral encoder: per-edge conv stack 5x80 ->32x40 ->64x20 ->64x10 -> mean.
// One 64-thread block per edge; activations staged in LDS (~13 KB).
// Writes xin[g*80 + 0..63] = sp*mask, [64..66] = normalized link_static*mask, rest 0.
// ---------------------------------------------------------------------------
__global__ void __launch_bounds__(64) k_spectral(
    const float* __restrict__ spectral,      // [BE,5,80]
    const float* __restrict__ link_static,   // [BE,3]
    const unsigned char* __restrict__ link_mask, // [BE]
    const float* __restrict__ c1w, const float* __restrict__ c1b,
    const float* __restrict__ c2w, const float* __restrict__ c2b,
    const float* __restrict__ c3w, const float* __restrict__ c3b,
    float* __restrict__ xin)                 // [BE,80]
{
  __shared__ float x0[5][84];    // input padded by 2 each side
  __shared__ float a1[32][44];   // conv1 out padded by 2 each side
  __shared__ float a2[64][22];   // conv2 out padded by 1 each side
  int g = blockIdx.x;
  int tid = threadIdx.x;
  const float* xg = spectral + (size_t)g * 5 * 80;

  for (int i = tid; i < 5 * 84; i += 64) {
    int c = i / 84, p = i - c * 84;
    x0[c][p] = (p >= 2 && p < 82) ? xg[c * 80 + p - 2] : 0.0f;
  }
  __syncthreads();

  // conv1: K=5, stride 2, pad 2 -> 32 x 40 ; padded index 2t+k
  for (int i = tid; i < 32 * 44; i += 64) {
    int c = i / 44, p = i - c * 44;
    float v = 0.0f;
    if (p >= 2 && p < 42) {
      int t = p - 2;
      float s = c1b[c];
      for (int ci = 0; ci < 5; ++ci) {
        const float* w = c1w + (c * 5 + ci) * 5;
        const float* xr = &x0[ci][2 * t];
        s += w[0]*xr[0] + w[1]*xr[1] + w[2]*xr[2] + w[3]*xr[3] + w[4]*xr[4];
      }
      v = fmaxf(s, 0.0f);
    }
    a1[c][p] = v;
  }
  __syncthreads();

  // conv2: K=5, stride 2, pad 2 -> 64 x 20
  for (int i = tid; i < 64 * 22; i += 64) {
    int c = i / 22, p = i - c * 22;
    float v = 0.0f;
    if (p >= 1 && p < 21) {
      int t = p - 1;
      float s = c2b[c];
      for (int ci = 0; ci < 32; ++ci) {
        const float* w = c2w + (c * 32 + ci) * 5;
        const float* xr = &a1[ci][2 * t];
        s += w[0]*xr[0] + w[1]*xr[1] + w[2]*xr[2] + w[3]*xr[3] + w[4]*xr[4];
      }
      v = fmaxf(s, 0.0f);
    }
    a2[c][p] = v;
  }
  __syncthreads();

  // conv3: K=3, stride 2, pad 1 -> 64 x 10, then mean over 10
  {
    int c = tid;   // 64 channels, 64 threads
    float m = 0.0f;
    for (int t = 0; t < 10; ++t) {
      float s = c3b[c];
      for (int ci = 0; ci < 64; ++ci) {
        const float* w = c3w + (c * 64 + ci) * 3;
        const float* xr = &a2[ci][2 * t];
        s += w[0]*xr[0] + w[1]*xr[1] + w[2]*xr[2];
      }
      m += fmaxf(s, 0.0f);
    }
    float lm = link_mask[g] ? 1.0f : 0.0f;
    xin[(size_t)g * 80 + c] = m * 0.1f * lm;
  }
  if (tid < 16) {
    float lm = link_mask[g] ? 1.0f : 0.0f;
    float v = 0.0f;
    if (tid < 3) {
      const float dv[3] = {200.0f, 10.0f, 10.0f};
      v = link_static[(size_t)g * 3 + tid] / dv[tid] * lm;
    }
    xin[(size_t)g * 80 + 64 + tid] = v;   // cols 64..79: ls then zero pad
  }
}

// ---------------------------------------------------------------------------
// GAT helpers
// ---------------------------------------------------------------------------
__global__ void k_gather_hv(const float* __restrict__ h, const float* __restrict__ e,
                            const int* __restrict__ edge_index, float* __restrict__ hv)
{
  size_t i = (size_t)blockIdx.x * blockDim.x + threadIdx.x;   // BE*256
  if (i >= (size_t)BEc * 256) return;
  int g = (int)(i >> 8);
  int d = (int)(i & 255);
  int b = g / Ec;
  if (d < 128) {
    int src = edge_index[(size_t)g * 2 + 0];
    hv[i] = h[((size_t)b * Nc + src) * 128 + d];
  } else {
    hv[i] = e[(size_t)g * 128 + (d - 128)];
  }
}

__global__ void k_attn(const float* __restrict__ Q, const float* __restrict__ Kf,
                       const float* __restrict__ e, const int* __restrict__ edge_index,
                       const unsigned char* __restrict__ link_mask,
                       const float* __restrict__ We_l, float* __restrict__ attn)
{
  int i = blockIdx.x * blockDim.x + threadIdx.x;   // BE*4
  if (i >= BEc * Hc) return;
  int g = i >> 2, hh = i & 3;
  int b = g / Ec;
  int src = edge_index[(size_t)g * 2 + 0];
  int dst = edge_index[(size_t)g * 2 + 1];
  const float* q = Q  + ((size_t)b * Nc + dst) * 128 + hh * 32;
  const float* k = Kf + ((size_t)b * Nc + src) * 128 + hh * 32;
  float s = 0.0f;
  for (int d = 0; d < 32; ++d) s += q[d] * k[d];
  s *= 0.17677669529663687f;   // 1/sqrt(32)
  const float* ee = e + (size_t)g * 128;
  const float* w = We_l + hh * 128;
  float t = 0.0f;
  for (int d = 0; d < 128; ++d) t += ee[d] * w[d];
  s += t;
  attn[i] = link_mask[g] ? s : -INFINITY;
}

__device__ inline void atomicMaxF(float* addr, float val) {
  // signed-int max for val>=0, unsigned-int min for val<0; init must be -inf
  if (val >= 0.0f) atomicMax((int*)addr, __float_as_int(val));
  else             atomicMin((unsigned int*)addr, __float_as_uint(val));
}

__global__ void k_scatter_max(const float* __restrict__ attn,
                              const int* __restrict__ edge_index, float* __restrict__ mx)
{
  int i = blockIdx.x * blockDim.x + threadIdx.x;   // BE*4
  if (i >= BEc * Hc) return;
  int g = i >> 2, hh = i & 3;
  int b = g / Ec;
  int dst = edge_index[(size_t)g * 2 + 1];
  atomicMaxF(&mx[((size_t)b * Nc + dst) * 4 + hh], attn[i]);
}

__global__ void k_exp_denom(float* __restrict__ attn, const int* __restrict__ edge_index,
                            const float* __restrict__ mx, float* __restrict__ denom)
{
  int i = blockIdx.x * blockDim.x + threadIdx.x;   // BE*4
  if (i >= BEc * Hc) return;
  int g = i >> 2, hh = i & 3;
  int b = g / Ec;
  int dst = edge_index[(size_t)g * 2 + 1];
  size_t idx = ((size_t)b * Nc + dst) * 4 + hh;
  float m = fmaxf(mx[idx], -1e6f);
  float ex = __expf(attn[i] - m);
  attn[i] = ex;
  atomicAdd(&denom[idx], ex);
}

__global__ void k_msg_scatter(const float* __restrict__ attn, const float* __restrict__ denom,
                              const float* __restrict__ V, const int* __restrict__ edge_index,
                              float* __restrict__ agg)
{
  size_t i = (size_t)blockIdx.x * blockDim.x + threadIdx.x;   // BE*128
  if (i >= (size_t)BEc * 128) return;
  int g = (int)(i >> 7);
  int d = (int)(i & 127);
  int b = g / Ec;
  int src = edge_index[(size_t)g * 2 + 0];
  int dst = edge_index[(size_t)g * 2 + 1];
  int hh = d >> 5;
  float w = attn[(size_t)g * 4 + hh] /
            fmaxf(denom[((size_t)b * Nc + dst) * 4 + hh], 1e-8f);
  float msg = w * V[(size_t)g * 128 + d];
  atomicAdd(&agg[((size_t)b * Nc + dst) * 128 + d], msg);
  atomicAdd(&agg[((size_t)b * Nc + src) * 128 + d], msg);
}

__global__ void __launch_bounds__(128) k_layernorm_mask(
    const float* __restrict__ tmp, const float* __restrict__ gW,
    const float* __restrict__ bW, const unsigned char* __restrict__ node_mask,
    float* __restrict__ out)
{
  __shared__ float s1[128], s2[128];
  int r = blockIdx.x;           // BN rows
  int tid = threadIdx.x;
  float v = tmp[(size_t)r * 128 + tid];
  s1[tid] = v; s2[tid] = v * v;
  __syncthreads();
  for (int off = 64; off > 0; off >>= 1) {
    if (tid < off) { s1[tid] += s1[tid + off]; s2[tid] += s2[tid + off]; }
    __syncthreads();
  }
  float mu = s1[0] * (1.0f / 128.0f);
  float var = s2[0] * (1.0f / 128.0f) - mu * mu;
  float y = (v - mu) * rsqrtf(var + 1e-5f) * gW[tid] + bW[tid];
  out[(size_t)r * 128 + tid] = y * (node_mask[r] ? 1.0f : 0.0f);
}

// ---------------------------------------------------------------------------
// Tails: LP MLP, masked means, final projection + batchnorm
// ---------------------------------------------------------------------------
__global__ void k_lp(const float* __restrict__ lp_features,
                     const float* __restrict__ l1w, const float* __restrict__ l1b,
                     const float* __restrict__ l2w, const float* __restrict__ l2b,
                     float* __restrict__ lp_out)
{
  int i = blockIdx.x * blockDim.x + threadIdx.x;   // B*LP rows
  if (i >= Bc * LPc) return;
  const float* x = lp_features + (size_t)i * 20;
  float hbuf[64];
  for (int j = 0; j < 64; ++j) {
    float s = l1b[j];
    for (int k = 0; k < 20; ++k) s += l1w[j * 20 + k] * x[k];
    hbuf[j] = fmaxf(s, 0.0f);
  }
  for (int j = 0; j < 32; ++j) {
    float s = l2b[j];
    for (int k = 0; k < 64; ++k) s += l2w[j * 64 + k] * hbuf[k];
    lp_out[(size_t)i * 32 + j] = s;
  }
}

__global__ void k_summary(const float* __restrict__ lp_out,
                          const unsigned char* __restrict__ lp_mask,
                          float* __restrict__ summary)
{
  int i = blockIdx.x * blockDim.x + threadIdx.x;   // B*32
  if (i >= Bc * 32) return;
  int b = i >> 5, j = i & 31;
  float s = 0.0f, c = 0.0f;
  for (int p = 0; p < LPc; ++p) {
    float m = lp_mask[b * LPc + p] ? 1.0f : 0.0f;
    s += lp_out[((size_t)b * LPc + p) * 32 + j] * m;
    c += m;
  }
  summary[i] = s / fmaxf(c, 1.0f);
}

__global__ void k_pool(const float* __restrict__ h,
                       const unsigned char* __restrict__ node_mask,
                       float* __restrict__ pooled)
{
  int i = blockIdx.x * blockDim.x + threadIdx.x;   // B*128
  if (i >= Bc * 128) return;
  int b = i >> 7, d = i & 127;
  float s = 0.0f, c = 0.0f;
  for (int n = 0; n < Nc; ++n) {
    float m = node_mask[b * Nc + n] ? 1.0f : 0.0f;
    s += h[((size_t)b * Nc + n) * 128 + d] * m;
    c += m;
  }
  pooled[i] = s / fmaxf(c, 1.0f);
}

__global__ void __launch_bounds__(1024) k_final(
    const float* __restrict__ pooled, const float* __restrict__ summary,
    const float* __restrict__ gfeat, const float* __restrict__ pw,
    const float* __restrict__ pb, const float* __restrict__ bn_g,
    const float* __restrict__ bn_b, float* __restrict__ out)
{
  __shared__ float zsh[8][128];
  __shared__ float mus[128], vars[128];
  int tid = threadIdx.x;
  int b = tid >> 7, j = tid & 127;
  float s = pb[j];
  const float* wrow = pw + (size_t)j * 168;   // 128 + 32 + 8
  for (int k = 0; k < 128; ++k) s += wrow[k] * pooled[b * 128 + k];
  for (int k = 0; k < 32; ++k)  s += wrow[128 + k] * summary[b * 32 + k];
  for (int k = 0; k < 8; ++k)   s += wrow[160 + k] * gfeat[b * 8 + k];
  zsh[b][j] = s;
  __syncthreads();
  if (tid < 128) {
    float m = 0.0f, q = 0.0f;
    for (int bb = 0; bb < 8; ++bb) { float z = zsh[bb][tid]; m += z; q += z * z; }
    m *= 0.125f; q = q * 0.125f - m * m;
    mus[tid] = m; vars[tid] = q;
  }
  __syncthreads();
  out[tid] = (zsh[b][j] - mus[j]) * rsqrtf(vars[j] + 1e-5f) * bn_g[j] + bn_b[j];
}

// ---------------------------------------------------------------------------
// Host orchestration
// ---------------------------------------------------------------------------
static inline int gdiv(long long n, int b) { return (int)((n + b - 1) / b); }

extern "C" void kernel_launch(void* const* d_in, const int* in_sizes, int n_in,
                              void* d_out, int out_size, void* d_ws, size_t ws_size,
                              hipStream_t stream) {
  const float* spectral       = (const float*)d_in[0];
  const float* link_static    = (const float*)d_in[1];
  const int*   edge_index     = (const int*)d_in[2];
  const unsigned char* link_mask = (const unsigned char*)d_in[3];
  const unsigned char* node_mask = (const unsigned char*)d_in[4];
  const float* node_feat      = (const float*)d_in[5];
  const float* lp_features    = (const float*)d_in[6];
  const unsigned char* lp_mask = (const unsigned char*)d_in[7];
  const float* gfeat          = (const float*)d_in[8];
  const float* c1w = (const float*)d_in[9];  const float* c1b = (const float*)d_in[10];
  const float* c2w = (const float*)d_in[11]; const float* c2b = (const float*)d_in[12];
  const float* c3w = (const float*)d_in[13]; const float* c3b = (const float*)d_in[14];
  const float* ew  = (const float*)d_in[15]; const float* eb  = (const float*)d_in[16];
  const float* Wq  = (const float*)d_in[17]; const float* Wk  = (const float*)d_in[18];
  const float* Wv  = (const float*)d_in[19]; const float* We  = (const float*)d_in[20];
  const float* Wo  = (const float*)d_in[21]; const float* bo  = (const float*)d_in[22];
  const float* ln_g = (const float*)d_in[23]; const float* ln_b = (const float*)d_in[24];
  const float* l1w = (const float*)d_in[25]; const float* l1b = (const float*)d_in[26];
  const float* l2w = (const float*)d_in[27]; const float* l2b = (const float*)d_in[28];
  const float* pw  = (const float*)d_in[29]; const float* pb  = (const float*)d_in[30];
  const float* bn_g = (const float*)d_in[31]; const float* bn_b = (const float*)d_in[32];

  float* ws = (float*)d_ws;
  size_t off = 0;
  auto alloc = [&](size_t n) { size_t o = off; off += n; return o; };

  float* e_buf  = ws + alloc((size_t)BEc * 128);
  float* h_a    = ws + alloc((size_t)BNc * 128);
  float* h_b    = ws + alloc((size_t)BNc * 128);
  float* Q_buf  = ws + alloc((size_t)BNc * 128);
  float* K_buf  = ws + alloc((size_t)BNc * 128);
  float* hv_buf = ws + alloc((size_t)BEc * 256);   // xin [BEc*80] aliases this region
  float* V_buf  = ws + alloc((size_t)BEc * 128);
  float* attn_b = ws + alloc((size_t)BEc * 4);
  float* mx_b   = ws + alloc((size_t)BNc * 4);
  float* den_b  = ws + alloc((size_t)BNc * 4);
  float* agg_b  = ws + alloc((size_t)BNc * 128);
  float* tmp_b  = ws + alloc((size_t)BNc * 128);
  float* WqT    = ws + alloc(128 * 128);           // also reused for ewT (80*128 fits)
  float* WkT    = ws + alloc(128 * 128);
  float* WvT    = ws + alloc(256 * 128);
  float* WoT    = ws + alloc(128 * 128);
  float* lp_out = ws + alloc((size_t)Bc * LPc * 32);
  float* summ   = ws + alloc(Bc * 32);
  float* pooled = ws + alloc(Bc * 128);
  float* xin    = hv_buf;                          // [BEc, 80]

  // 1) spectral conv stack + edge-feature assembly
  k_spectral<<<BEc, 64, 0, stream>>>(spectral, link_static, link_mask,
                                     c1w, c1b, c2w, c2b, c3w, c3b, xin);

  // 2) edge projection: e = relu(xin[.,:67] @ ew.T + eb), K padded 67->80
  k_transpose_pad<<<gdiv(80 * 128, 256), 256, 0, stream>>>(ew, WqT, 128, 67, 80);
  k_wmma_gemm<<<BEc / 16, 256, 0, stream>>>(
      xin, 80, WqT, eb, nullptr, e_buf, 80, 1);

  // 3) h = node_feat
  hipMemcpyAsync(h_a, node_feat, (size_t)BNc * 128 * sizeof(float),
                 hipMemcpyDeviceToDevice, stream);

  float* h_cur = h_a;
  float* h_nxt = h_b;
  for (int l = 0; l < NLc; ++l) {
    const float* Wq_l = Wq + (size_t)l * 128 * 128;
    const float* Wk_l = Wk + (size_t)l * 128 * 128;
    const float* Wv_l = Wv + (size_t)l * 128 * 256;
    const float* We_l = We + (size_t)l * 4 * 128;
    const float* Wo_l = Wo + (size_t)l * 128 * 128;

    k_transpose_pad<<<gdiv(128 * 128, 256), 256, 0, stream>>>(Wq_l, WqT, 128, 128, 128);
    k_transpose_pad<<<gdiv(128 * 128, 256), 256, 0, stream>>>(Wk_l, WkT, 128, 128, 128);
    k_transpose_pad<<<gdiv(256 * 128, 256), 256, 0, stream>>>(Wv_l, WvT, 128, 256, 256);
    k_transpose_pad<<<gdiv(128 * 128, 256), 256, 0, stream>>>(Wo_l, WoT, 128, 128, 128);

    // Q, K projections (WMMA)
    k_wmma_gemm<<<BNc / 16, 256, 0, stream>>>(
        h_cur, 128, WqT, nullptr, nullptr, Q_buf, 128, 0);
    k_wmma_gemm<<<BNc / 16, 256, 0, stream>>>(
        h_cur, 128, WkT, nullptr, nullptr, K_buf, 128, 0);

    // V = [h[src], e] @ Wv.T (gather then WMMA, K=256)
    k_gather_hv<<<gdiv((long long)BEc * 256, 256), 256, 0, stream>>>(
        h_cur, e_buf, edge_index, hv_buf);
    k_wmma_gemm<<<BEc / 16, 256, 0, stream>>>(
        hv_buf, 256, WvT, nullptr, nullptr, V_buf, 256, 0);

    // attention logits
    k_attn<<<gdiv(BEc * 4, 256), 256, 0, stream>>>(
        Q_buf, K_buf, e_buf, edge_index, link_mask, We_l, attn_b);

    // scatter softmax over incoming edges
    k_fill<<<gdiv(BNc * 4, 256), 256, 0, stream>>>(mx_b, -INFINITY, (size_t)BNc * 4);
    k_fill<<<gdiv(BNc * 4, 256), 256, 0, stream>>>(den_b, 0.0f, (size_t)BNc * 4);
    k_fill<<<gdiv((long long)BNc * 128, 256), 256, 0, stream>>>(agg_b, 0.0f, (size_t)BNc * 128);
    k_scatter_max<<<gdiv(BEc * 4, 256), 256, 0, stream>>>(attn_b, edge_index, mx_b);
    k_exp_denom<<<gdiv(BEc * 4, 256), 256, 0, stream>>>(attn_b, edge_index, mx_b, den_b);
    k_msg_scatter<<<gdiv((long long)BEc * 128, 256), 256, 0, stream>>>(
        attn_b, den_b, V_buf, edge_index, agg_b);

    // out = layernorm(h + agg @ Wo.T + bo) * node_mask
    k_wmma_gemm<<<BNc / 16, 256, 0, stream>>>(
        agg_b, 128, WoT, bo + l * 128, h_cur, tmp_b, 128, 0);
    k_layernorm_mask<<<BNc, 128, 0, stream>>>(
        tmp_b, ln_g + l * 128, ln_b + l * 128, node_mask, h_nxt);

    float* t = h_cur; h_cur = h_nxt; h_nxt = t;
  }

  // tails
  k_pool<<<gdiv(Bc * 128, 256), 256, 0, stream>>>(h_cur, node_mask, pooled);
  k_lp<<<gdiv(Bc * LPc, 128), 128, 0, stream>>>(lp_features, l1w, l1b, l2w, l2b, lp_out);
  k_summary<<<1, 256, 0, stream>>>(lp_out, lp_mask, summ);
  k_final<<<1, 1024, 0, stream>>>(pooled, summ, gfeat, pw, pb, bn_g, bn_b, (float*)d_out);
}